// CT_37821482009425
// MI455X (gfx1250) — compile-verified
//
#include <hip/hip_runtime.h>
#include <stdint.h>

#ifndef __has_builtin
#define __has_builtin(x) 0
#endif

// Problem constants (from reference)
#define B_    1024
#define H_    224
#define W_    224
#define BETA_ 0.95f

// Tiling: 64 rows / 64 threads per block -> 58,368 B LDS (< 64 KB cap),
// ~5 blocks resident per 320 KB WGP for cross-block DMA/scan overlap.
#define ROWS  64
#define BLK   64
#define LSTR  228     // padded LDS row stride (floats): 912B = 16B-aligned, <=2-way bank conflict

// Typed address-space pointers matching the async builtin signatures:
//   b128: (int4 AS1*, int4 AS3*, Ii offset, Ii cpol)
//   b32 : (int  AS1*, int  AS3*, Ii offset, Ii cpol)
typedef int v4i __attribute__((ext_vector_type(4)));
typedef __attribute__((address_space(1))) v4i g_v4i;
typedef __attribute__((address_space(3))) v4i l_v4i;
typedef __attribute__((address_space(1))) int g_i32;
typedef __attribute__((address_space(3))) int l_i32;

// ---- CDNA5 async global<->LDS data movers (ASYNCcnt-tracked) -------------
__device__ __forceinline__ void async_g2l_b128(const float* g, float* l) {
#if __has_builtin(__builtin_amdgcn_global_load_async_to_lds_b128)
    __builtin_amdgcn_global_load_async_to_lds_b128((g_v4i*)g, (l_v4i*)l, 0, 0);
#else
    *(float4*)l = *(const float4*)g;   // fallback: VGPR round-trip
#endif
}

__device__ __forceinline__ void async_l2g_b32(float* g, const float* l) {
#if __has_builtin(__builtin_amdgcn_global_store_async_from_lds_b32)
    __builtin_amdgcn_global_store_async_from_lds_b32((g_i32*)g, (l_i32*)l, 0, 0);
#else
    *g = *l;                            // fallback: VGPR round-trip
#endif
}

__device__ __forceinline__ void wait_async0() {
#if __has_builtin(__builtin_amdgcn_s_wait_asynccnt)
    __builtin_amdgcn_s_wait_asynccnt(0);
#else
    asm volatile("s_wait_asynccnt 0" ::: "memory");
#endif
}

// ---------------------------------------------------------------------------
// LIF scan: one thread owns one (b,h) row; roll folded into LDS read index.
// ---------------------------------------------------------------------------
__global__ __launch_bounds__(BLK) void lif_scan_kernel(
    const float* __restrict__ inp,     // (B, H, W) f32
    const float* __restrict__ tparam,  // (1,) threshold
    const int*   __restrict__ rollp,   // (1,) roll amount
    float*       __restrict__ out)     // (B, 1, H, W) f32
{
    extern __shared__ float tile[];    // ROWS * LSTR floats (~57 KB)

    const int  tid     = threadIdx.x;
    const long rowBase = (long)blockIdx.x * ROWS;   // flat row = b*H + h

    float T = tparam[0];
    T = fminf(fmaxf(T, 1.0f), 5.0f);
    int shift = rollp[0] % W_; if (shift < 0) shift += W_;

    // ---- Phase 1: async DMA global -> LDS, 16B granules, fully coalesced ----
    {
        const float* gbase = inp + rowBase * W_;
        const int NG = ROWS * (W_ / 4);            // 3584 b128 granules
        for (int g = tid; g < NG; g += BLK) {      // 56 full iterations
            int row = g / (W_ / 4);
            int c4  = (g - row * (W_ / 4)) * 4;    // column (input order)
            async_g2l_b128(gbase + row * W_ + c4, &tile[row * LSTR + c4]);
        }
    }
    wait_async0();        // each wave drains its own ASYNCcnt
    __syncthreads();      // then cross-wave visibility

    // ---- Phase 2: sequential leaky integrate-and-fire scan per row ----
    // Scan step s reads LDS column c = (s - shift) mod W; split into two
    // wrap-free segments so the inner loop has no modulo on the hot chain.
    // Each LDS slot is read exactly once, then overwritten with its spike
    // (the spike for output time s = (c + shift) mod W).
    {
        float  mem  = 0.0f;
        float* lrow = &tile[tid * LSTR];
        int c0 = W_ - shift; if (c0 >= W_) c0 -= W_;   // column at s=0

        #pragma unroll 4
        for (int c = c0; c < W_; ++c) {
            float x   = lrow[c];
            float acc = __builtin_fmaf(BETA_, mem, x);
            mem = (mem > T) ? acc - T : acc;           // subtract-reset
            lrow[c] = (mem > T) ? 1.0f : 0.0f;         // spike, in place
        }
        #pragma unroll 4
        for (int c = 0; c < c0; ++c) {
            float x   = lrow[c];
            float acc = __builtin_fmaf(BETA_, mem, x);
            mem = (mem > T) ? acc - T : acc;
            lrow[c] = (mem > T) ? 1.0f : 0.0f;
        }
    }
    __syncthreads();

    // ---- Phase 3: async DMA LDS -> global, un-rotating positions ----
    {
        float* obase = out + rowBase * W_;
        const int NE = ROWS * W_;                  // 14336 elements
        for (int e = tid; e < NE; e += BLK) {      // 224 full iterations
            int row = e / W_;
            int c   = e - row * W_;                // LDS slot (input order)
            int s   = c + shift; if (s >= W_) s -= W_;   // output time index
            async_l2g_b32(obase + row * W_ + s, &tile[row * LSTR + c]);
        }
    }
    wait_async0();   // S_ENDPGM also implicitly waits all counters
}

// ---------------------------------------------------------------------------
extern "C" void kernel_launch(void* const* d_in, const int* in_sizes, int n_in,
                              void* d_out, int out_size, void* d_ws, size_t ws_size,
                              hipStream_t stream) {
    const float* inp  = (const float*)d_in[0];
    const float* t    = (const float*)d_in[1];
    const int*   roll = (const int*)d_in[2];
    float*       out  = (float*)d_out;

    const int totalRows = B_ * H_;                 // 229376 = 3584 * 64
    dim3 grid(totalRows / ROWS);
    dim3 block(BLK);
    size_t ldsBytes = (size_t)ROWS * LSTR * sizeof(float);   // 58,368 B
    lif_scan_kernel<<<grid, block, ldsBytes, stream>>>(inp, t, roll, out);
}